// SDSG7_3496103379547
// MI455X (gfx1250) — compile-verified
//
#include <hip/hip_runtime.h>
#include <hip/hip_bf16.h>
#include <math.h>

typedef float v2f __attribute__((ext_vector_type(2)));
typedef float v8f __attribute__((ext_vector_type(8)));

#define HID 32

// ---------------------------------------------------------------------------
// Fill a float buffer with a constant (used for deg init = 1.0 (self loop) and
// zeroing the aggregation accumulator).
// ---------------------------------------------------------------------------
__global__ void k_fill(float* __restrict__ p, long long n, float v) {
    long long i = (long long)blockIdx.x * blockDim.x + threadIdx.x;
    if (i < n) p[i] = v;
}

// deg[col[e]] += 1  (destination-degree with self loops already counted via init=1)
__global__ void k_deg(const int* __restrict__ ei, float* __restrict__ deg,
                      int E) {
    int e = blockIdx.x * blockDim.x + threadIdx.x;
    if (e >= E) return;
    int c = ei[E + e];
    atomicAdd(&deg[c], 1.0f);
}

// dis = deg > 0 ? rsqrt(deg) : 0   (in place)
__global__ void k_rsqrt(float* __restrict__ deg, int n) {
    int i = blockIdx.x * blockDim.x + threadIdx.x;
    if (i >= n) return;
    float d = deg[i];
    deg[i] = d > 0.0f ? rsqrtf(d) : 0.0f;
}

// ---------------------------------------------------------------------------
// WMMA fp32 GEMM: out[n x 32] = A[n x K] @ W[K x 32] + bias, optional ReLU.
// One wave handles a 16-node tile -> two 16x16 f32 accumulators via
// V_WMMA_F32_16X16X4_F32, stepping K by 4.
// Operand layouts (CDNA5 ISA 7.12.2, 32-bit f32):
//   A (16x4):  lane l, vgpr j -> A[M = l%16][K = kbase + j + 2*(l/16)]
//   B (4x16):  lane l, vgpr j -> B[K = kbase + j + 2*(l/16)][N = l%16]
//   C/D 16x16: lane l, vgpr v -> C[M = v + 8*(l/16)][N = l%16]
// ---------------------------------------------------------------------------
__global__ void k_gemm_wmma(const float* __restrict__ A, const float* __restrict__ W,
                            const float* __restrict__ bias, float* __restrict__ out,
                            int n, int K, int relu) {
    int tid  = blockIdx.x * blockDim.x + threadIdx.x;
    int wave = tid >> 5;
    int lane = tid & 31;
    int node0 = wave * 16;
    if (node0 >= n) return;           // uniform per wave -> EXEC stays all-1s

    int l16 = lane & 15;
    int hi  = lane >> 4;              // 0 or 1 (lane half)
    int arow = node0 + l16;
    if (arow > n - 1) arow = n - 1;   // clamp tail reads (stores guarded below)

    v8f c0 = {};
    v8f c1 = {};
    for (int k = 0; k < K; k += 4) {
        int ka = k + 2 * hi;
        v2f a;
        a.x = A[(size_t)arow * K + ka + 0];
        a.y = A[(size_t)arow * K + ka + 1];
        v2f b0, b1;
        b0.x = W[(ka + 0) * HID + l16];
        b0.y = W[(ka + 1) * HID + l16];
        b1.x = W[(ka + 0) * HID + 16 + l16];
        b1.y = W[(ka + 1) * HID + 16 + l16];
        c0 = __builtin_amdgcn_wmma_f32_16x16x4_f32(false, a, false, b0,
                                                   (short)0, c0, false, false);
        c1 = __builtin_amdgcn_wmma_f32_16x16x4_f32(false, a, false, b1,
                                                   (short)0, c1, false, false);
    }

    float bias0 = bias[l16];
    float bias1 = bias[16 + l16];
    #pragma unroll
    for (int v = 0; v < 8; ++v) {
        int node = node0 + v + hi * 8;
        if (node >= n) continue;
        float r0 = c0[v] + bias0;
        float r1 = c1[v] + bias1;
        if (relu) {
            r0 = r0 > 0.0f ? r0 : 0.0f;
            r1 = r1 > 0.0f ? r1 : 0.0f;
        }
        out[(size_t)node * HID + l16]      = r0;
        out[(size_t)node * HID + 16 + l16] = r1;
    }
}

// In-place per-row min-max normalization to [-1,1] over 32 features.
__global__ void k_mynorm(float* __restrict__ x, int n) {
    int i = blockIdx.x * blockDim.x + threadIdx.x;
    if (i >= n) return;
    float* row = x + (size_t)i * HID;
    float vals[HID];
    float mn = 1e30f, mx = -1e30f;
    #pragma unroll
    for (int j = 0; j < HID; ++j) {
        float v = row[j];
        vals[j] = v;
        mn = fminf(mn, v);
        mx = fmaxf(mx, v);
    }
    float inv = 2.0f / (mx - mn + 1e-8f);
    #pragma unroll
    for (int j = 0; j < HID; ++j) row[j] = (vals[j] - mn) * inv - 1.0f;
}

// Edge aggregation: one wave per edge, lane = feature channel.
// agg[col] += x[row] * dis[row] * dis[col]
__global__ void k_edge_agg(const int* __restrict__ ei,
                           const float* __restrict__ dis,
                           const float* __restrict__ xin,
                           float* __restrict__ agg, int E) {
    long long gid = (long long)blockIdx.x * blockDim.x + threadIdx.x;
    int e = (int)(gid >> 5);
    int lane = (int)(gid & 31);
    if (e >= E) return;
    int r = ei[e];
    int c = ei[E + e];
    float nv = dis[r] * dis[c];
    float v = xin[(size_t)r * HID + lane] * nv;
    atomicAdd(&agg[(size_t)c * HID + lane], v);
}

// Self-loop contribution (runs after edge aggregation; plain add, no race).
__global__ void k_self(const float* __restrict__ xin, const float* __restrict__ dis,
                       float* __restrict__ agg, long long total) {
    long long i = (long long)blockIdx.x * blockDim.x + threadIdx.x;
    if (i >= total) return;
    long long node = i >> 5;
    float d = dis[node];
    agg[i] += xin[i] * d * d;
}

// ---------------------------------------------------------------------------
// Final stage: wave per node. 32 lanes = 32 channels. Compute mynorm of each
// of the 7 layer rows with shfl_xor min/max trees, form the 224-wide concat
// row implicitly, then reduce 16 outputs of fc7 across the wave.
// ---------------------------------------------------------------------------
__global__ void k_fc7(const float* __restrict__ xs,  // 7 contiguous [n x 32] blocks
                      const float* __restrict__ w,   // [224 x 16]
                      const float* __restrict__ b,   // [16]
                      float* __restrict__ out, int n) {
    int tid  = blockIdx.x * blockDim.x + threadIdx.x;
    int node = tid >> 5;
    int lane = tid & 31;
    if (node >= n) return;

    float raw[7], m[7];
    #pragma unroll
    for (int k = 0; k < 7; ++k) {
        float v = xs[(size_t)k * n * HID + (size_t)node * HID + lane];
        raw[k] = v;
        float mn = v, mx = v;
        #pragma unroll
        for (int off = 16; off >= 1; off >>= 1) {
            mn = fminf(mn, __shfl_xor(mn, off, 32));
            mx = fmaxf(mx, __shfl_xor(mx, off, 32));
        }
        m[k] = 2.0f * (v - mn) / (mx - mn + 1e-8f) - 1.0f;
    }

    float xx[7];
    xx[0] = raw[0];
    xx[1] = raw[1];
    xx[2] = m[2] - m[0];
    xx[3] = m[3] - m[1];
    xx[4] = m[4] - m[2];
    xx[5] = m[5] - m[3];
    xx[6] = m[6] - m[4];

    #pragma unroll 4
    for (int o = 0; o < 16; ++o) {
        float p = 0.0f;
        #pragma unroll
        for (int j = 0; j < 7; ++j)
            p += xx[j] * w[(j * 32 + lane) * 16 + o];
        #pragma unroll
        for (int off = 16; off >= 1; off >>= 1)
            p += __shfl_xor(p, off, 32);
        if (lane == 0) out[(size_t)node * 16 + o] = p + b[o];
    }
}

extern "C" void kernel_launch(void* const* d_in, const int* in_sizes, int n_in,
                              void* d_out, int out_size, void* d_ws, size_t ws_size,
                              hipStream_t stream) {
    const float* x      = (const float*)d_in[0];
    const int*   ei     = (const int*)d_in[1];     // harness delivers integers as int32
    const float* fc1_w  = (const float*)d_in[2];
    const float* fc1_b  = (const float*)d_in[3];
    const float* conv_w = (const float*)d_in[4];
    const float* conv_b = (const float*)d_in[5];
    const float* fc7_w  = (const float*)d_in[6];
    const float* fc7_b  = (const float*)d_in[7];
    float*       out    = (float*)d_out;

    const int n = in_sizes[0] / 128;   // 100000
    const int E = in_sizes[1] / 2;     // 3200000

    // Workspace layout
    char*  ws      = (char*)d_ws;
    float* xs      = (float*)ws;                       // 7 * n * 32 floats
    size_t xsElems = (size_t)7 * n * HID;
    float* deg     = xs + xsElems;                     // n floats (deg -> dis)
    float* agg     = deg + n;                          // n * 32 floats
    (void)ws_size; (void)n_in; (void)out_size;

    const int B = 256;
    long long nh = (long long)n * HID;

    // gcn_norm: deg (init 1.0 for self loops) -> dis = rsqrt(deg)
    k_fill<<<(unsigned)((n + B - 1) / B), B, 0, stream>>>(deg, n, 1.0f);
    k_deg<<<(unsigned)((E + B - 1) / B), B, 0, stream>>>(ei, deg, E);
    k_rsqrt<<<(unsigned)((n + B - 1) / B), B, 0, stream>>>(deg, n);

    // x0 = mynorm(relu(x @ fc1_w + fc1_b))
    {
        long long waves = (n + 15) / 16;
        unsigned grid = (unsigned)((waves * 32 + B - 1) / B);
        k_gemm_wmma<<<grid, B, 0, stream>>>(x, fc1_w, fc1_b, xs, n, 128, 1);
        k_mynorm<<<(unsigned)((n + B - 1) / B), B, 0, stream>>>(xs, n);
    }

    // 6 SGConv layers
    for (int i = 0; i < 6; ++i) {
        float* xin  = xs + (size_t)i * n * HID;
        float* xout = xs + (size_t)(i + 1) * n * HID;

        k_fill<<<(unsigned)((nh + B - 1) / B), B, 0, stream>>>(agg, nh, 0.0f);
        k_edge_agg<<<(unsigned)(((long long)E * 32 + B - 1) / B), B, 0, stream>>>(ei, deg, xin, agg, E);
        k_self<<<(unsigned)((nh + B - 1) / B), B, 0, stream>>>(xin, deg, agg, nh);

        long long waves = (n + 15) / 16;
        unsigned grid = (unsigned)((waves * 32 + B - 1) / B);
        k_gemm_wmma<<<grid, B, 0, stream>>>(agg, conv_w + (size_t)i * HID * HID,
                                            conv_b + (size_t)i * HID, xout, n, HID, 0);
    }

    // Final concat + fc7
    k_fc7<<<(unsigned)((nh + B - 1) / B), B, 0, stream>>>(xs, fc7_w, fc7_b, out, n);
}